// _ZExpectationModule_31662498906419
// MI455X (gfx1250) — compile-verified
//
#include <hip/hip_runtime.h>

// ======================================================================
// 23-qubit Z-expectation, algebraically reduced:
//   E = (1/S) * sum over pairs (chi[g], chi[g^0x600000]) of quadform(M)
//   chi = (tensor_prod of 23 layer-1 2x2 gates) applied to raw input state
//   M   = u0^dag Z u0  (only layer-2 gate that survives), S = ||psi||^2
// Gates applied 4 bit-positions at a time as 16x16 complex matmuls via
// V_WMMA_F32_16X16X4_F32 (6 passes over the 64MB state, L2-resident).
// ======================================================================

#define NQ   23
#define DIMQ (1u << NQ)

typedef __attribute__((ext_vector_type(2))) float v2f;
typedef __attribute__((ext_vector_type(8))) float v8f;

// ws float-offset layout
#define WS_S     0          // ||psi||^2 accumulator
#define WS_E     1          // unnormalized expectation accumulator
#define WS_M     2          // m00, m11, m01r, m01i  (2..5)
#define WS_W     16         // 6 groups * (256 re + 256 im) floats
#define WS_CHI_RE 4096u
#define WS_CHI_IM (4096u + DIMQ)

// V_q = RZ(tz) * RY(ty)  (reference applies RY then RZ)
__device__ inline void gate_v(float ty, float tz, float vr[2][2], float vi[2][2]) {
    float cy = __cosf(ty * 0.5f), sy = __sinf(ty * 0.5f);
    float cz = __cosf(tz * 0.5f), sz = __sinf(tz * 0.5f);
    vr[0][0] =  cz * cy; vi[0][0] = -sz * cy;
    vr[0][1] = -cz * sy; vi[0][1] =  sz * sy;
    vr[1][0] =  cz * sy; vi[1][0] =  sz * sy;
    vr[1][1] =  cz * cy; vi[1][1] =  sz * cy;
}

// Build the six 16x16 group matrices + measurement matrix M + zero accumulators.
__global__ void prep_kernel(const float* __restrict__ thetas, float* __restrict__ ws) {
    int t = threadIdx.x;               // 0..255 : (m = t>>4, k = t&15)
    if (t == 0) {
        ws[WS_S] = 0.0f;
        ws[WS_E] = 0.0f;
        // u = RY(th47) * RX(th46); M = u^dag Z u
        float c  = __cosf(thetas[46] * 0.5f), s  = __sinf(thetas[46] * 0.5f);
        float cy = __cosf(thetas[47] * 0.5f), sy = __sinf(thetas[47] * 0.5f);
        float u00r =  cy * c, u00i =  sy * s;
        float u01r = -sy * c, u01i = -cy * s;
        float u10r =  sy * c, u10i = -cy * s;
        float u11r =  cy * c, u11i = -sy * s;
        ws[WS_M + 0] = (u00r*u00r + u00i*u00i) - (u10r*u10r + u10i*u10i);       // m00
        ws[WS_M + 1] = (u01r*u01r + u01i*u01i) - (u11r*u11r + u11i*u11i);       // m11
        ws[WS_M + 2] = (u00r*u01r + u00i*u01i) - (u10r*u11r + u10i*u11i);       // Re m01
        ws[WS_M + 3] = (u00r*u01i - u00i*u01r) - (u10r*u11i - u10i*u11r);       // Im m01
    }
    int m = t >> 4, k = t & 15;
    for (int g = 0; g < 6; ++g) {
        float wr = 1.0f, wi = 0.0f;
        #pragma unroll
        for (int j = 0; j < 4; ++j) {
            int bm = (m >> j) & 1, bk = (k >> j) & 1;
            float fr, fi;
            if (g == 5 && j == 0) {            // identity factor on bit-position 19
                fr = (bm == bk) ? 1.0f : 0.0f; fi = 0.0f;
            } else {
                int p = (g < 5) ? (4 * g + j) : (19 + j);   // bit position
                int q = 22 - p;                              // qubit (0 = MSB)
                float vr[2][2], vi[2][2];
                gate_v(thetas[2 * q], thetas[2 * q + 1], vr, vi);
                fr = vr[bm][bk]; fi = vi[bm][bk];
            }
            float nr = wr * fr - wi * fi;
            float ni = wr * fi + wi * fr;
            wr = nr; wi = ni;
        }
        ws[WS_W + g * 512 + t]       = wr;
        ws[WS_W + g * 512 + 256 + t] = wi;
    }
}

// Apply one 16x16 complex matrix along a 4-bit axis at `shift`.
// mode 0: read inputs, write chi, accumulate S
// mode 1: in-place on chi
// mode 2: read chi, no write, accumulate E (axis must be bits 19..22)
__global__ __launch_bounds__(256) void apply_kernel(
    const float* src_re, const float* src_im,
    float* dst_re, float* dst_im,
    const float* __restrict__ wmat, float* ws_acc,
    int shift, int mode)
{
    __shared__ float lwr[256], lwi[256];
    const int tid = threadIdx.x;
    lwr[tid] = wmat[tid];
    lwi[tid] = wmat[256 + tid];
    __syncthreads();

    const int lane = tid & 31;
    const int h    = lane >> 4;      // wave half
    const int n    = lane & 15;      // fiber column / A row
    const int gwave  = blockIdx.x * (blockDim.x >> 5) + (tid >> 5);
    const int nwaves = gridDim.x * (blockDim.x >> 5);
    const unsigned nTiles = DIMQ >> 8;
    const unsigned loMask = (1u << shift) - 1u;

    // A fragments (W): ISA 16x4 f32 layout -> lane M = n, VGPR v holds K = 4c + v + 2h
    v2f ar[4], ai[4];
    #pragma unroll
    for (int c = 0; c < 4; ++c) {
        int k0 = 4 * c + 2 * h;
        ar[c].x = lwr[n * 16 + k0];     ar[c].y = lwr[n * 16 + k0 + 1];
        ai[c].x = lwi[n * 16 + k0];     ai[c].y = lwi[n * 16 + k0 + 1];
    }

    float m00 = 0.f, m11 = 0.f, m01r = 0.f, m01i = 0.f;
    if (mode == 2) {
        m00 = ws_acc[WS_M + 0]; m11 = ws_acc[WS_M + 1];
        m01r = ws_acc[WS_M + 2]; m01i = ws_acc[WS_M + 3];
    }

    float sAcc = 0.0f, eAcc = 0.0f;

    for (unsigned tile = (unsigned)gwave; tile < nTiles; tile += (unsigned)nwaves) {
        unsigned F   = (tile << 4) + (unsigned)n;          // fiber id
        unsigned flo = F & loMask;
        unsigned fhi = F >> shift;
        unsigned base = (fhi << (shift + 4)) | flo;

        // B fragments (state): VGPR v at lane half h holds gate row 4c + 2v + h
        v2f br[4], bi[4];
        #pragma unroll
        for (int c = 0; c < 4; ++c) {
            unsigned g0 = (unsigned)(4 * c + h);
            unsigned i0 = base + (g0 << shift);
            unsigned i1 = base + ((g0 + 2u) << shift);
            br[c].x = src_re[i0]; br[c].y = src_re[i1];
            bi[c].x = src_im[i0]; bi[c].y = src_im[i1];
            if (mode == 0)
                sAcc += br[c].x * br[c].x + bi[c].x * bi[c].x
                      + br[c].y * br[c].y + bi[c].y * bi[c].y;
        }

        // Complex 16x16 x 16x16: Dr = Wr*Br - Wi*Bi ; Di = Wr*Bi + Wi*Br
        v8f accRR = {}, accII = {}, accRI = {}, accIR = {};
        #pragma unroll
        for (int c = 0; c < 4; ++c) {
            accRR = __builtin_amdgcn_wmma_f32_16x16x4_f32(false, ar[c], false, br[c], (short)0, accRR, false, false);
            accII = __builtin_amdgcn_wmma_f32_16x16x4_f32(false, ai[c], false, bi[c], (short)0, accII, false, false);
            accRI = __builtin_amdgcn_wmma_f32_16x16x4_f32(false, ar[c], false, bi[c], (short)0, accRI, false, false);
            accIR = __builtin_amdgcn_wmma_f32_16x16x4_f32(false, ai[c], false, br[c], (short)0, accIR, false, false);
        }
        v8f dr = accRR - accII;
        v8f di = accRI + accIR;

        if (mode != 2) {
            // D layout: lane holds (M = r + 8h, N = n)
            #pragma unroll
            for (int r = 0; r < 8; ++r) {
                unsigned Mo = (unsigned)(r + 8 * h);
                unsigned io = base + (Mo << shift);
                dst_re[io] = dr[r];
                dst_im[io] = di[r];
            }
        } else {
            // Pair a = (M=r, lanes<16, bit22=0) with b = (M = r^12) living at
            // lane^16, register r^4.  E += m00|a|^2 + m11|b|^2 + 2Re(m01 a* b)
            #pragma unroll
            for (int r = 0; r < 8; ++r) {
                float pr = __shfl_xor(dr[r ^ 4], 16);
                float pi = __shfl_xor(di[r ^ 4], 16);
                if (h == 0) {
                    float arr = dr[r], aii = di[r];
                    eAcc += m00 * (arr * arr + aii * aii)
                          + m11 * (pr * pr + pi * pi)
                          + 2.0f * (m01r * (arr * pr + aii * pi)
                                  - m01i * (arr * pi - aii * pr));
                }
            }
        }
    }

    if (mode == 0) {
        #pragma unroll
        for (int off = 16; off; off >>= 1) sAcc += __shfl_xor(sAcc, off);
        if (lane == 0) atomicAdd(&ws_acc[WS_S], sAcc);
    }
    if (mode == 2) {
        #pragma unroll
        for (int off = 16; off; off >>= 1) eAcc += __shfl_xor(eAcc, off);
        if (lane == 0) atomicAdd(&ws_acc[WS_E], eAcc);
    }
}

__global__ void finish_kernel(const float* __restrict__ ws, float* __restrict__ out) {
    out[0] = ws[WS_E] / ws[WS_S];
}

extern "C" void kernel_launch(void* const* d_in, const int* in_sizes, int n_in,
                              void* d_out, int out_size, void* d_ws, size_t ws_size,
                              hipStream_t stream) {
    const float* thetas = (const float*)d_in[0];
    const float* sre    = (const float*)d_in[1];
    const float* sim    = (const float*)d_in[2];
    float* ws     = (float*)d_ws;
    float* out    = (float*)d_out;
    float* chi_re = ws + WS_CHI_RE;
    float* chi_im = ws + WS_CHI_IM;

    prep_kernel<<<1, 256, 0, stream>>>(thetas, ws);

    const int blocks = 1024, threads = 256;
    // pass 0: axis bits 0..3 (qubits 19..22), read inputs, fuse ||psi||^2
    apply_kernel<<<blocks, threads, 0, stream>>>(sre, sim, chi_re, chi_im,
                                                 ws + WS_W + 0 * 512, ws, 0, 0);
    // passes 1..4: axis bits 4..19, in place
    for (int g = 1; g <= 4; ++g)
        apply_kernel<<<blocks, threads, 0, stream>>>(chi_re, chi_im, chi_re, chi_im,
                                                     ws + WS_W + g * 512, ws, 4 * g, 1);
    // pass 5: axis bits 19..22 (identity on 19), fuse <Z0> pair reduction
    apply_kernel<<<blocks, threads, 0, stream>>>(chi_re, chi_im, chi_re, chi_im,
                                                 ws + WS_W + 5 * 512, ws, 19, 2);

    finish_kernel<<<1, 1, 0, stream>>>(ws, out);
}